// LightGCN_25881472925719
// MI455X (gfx1250) — compile-verified
//
#include <hip/hip_runtime.h>
#include <cstdint>

#define EMB  128   // embedding dim (fixed by reference)
#define TILE 256   // edges per block
#define BLK  256   // threads per block (8 wave32s)

typedef __attribute__((ext_vector_type(4))) uint32_t u32x4;
typedef __attribute__((ext_vector_type(8))) uint32_t u32x8;

static inline int cdiv(long a, long b) { return (int)((a + b - 1) / b); }

// Hardware f32 atomic add (global_atomic_add_f32); no return needed.
__device__ __forceinline__ void atomic_fadd(float* p, float v) {
  unsafeAtomicAdd(p, v);
}

// ---------------------------------------------------------------------------
// TDM: 1D strip load Global -> LDS via tensor_load_to_lds (2-group D#).
//   lds_off     : byte offset of destination within this workgroup's LDS
//   gptr        : global source (4-byte elements)
//   elems_avail : valid elements from gptr (tensor_dim0); OOB reads -> 0
//   tile_elems  : elements to transfer (tile_dim0)
// All inputs are uniform, so the descriptor lives in SGPRs.
// ---------------------------------------------------------------------------
__device__ __forceinline__ void tdm_load_1d(uint32_t lds_off, const void* gptr,
                                            uint32_t elems_avail, uint32_t tile_elems) {
  const uint64_t ga = (uint64_t)(uintptr_t)gptr;

  u32x4 g0;
  g0[0] = 1u;                                        // count=1, user desc, no gather
  g0[1] = lds_off;                                   // lds_addr (bytes)
  g0[2] = (uint32_t)ga;                              // global_addr[31:0]
  g0[3] = ((uint32_t)(ga >> 32) & 0x01FFFFFFu)       // global_addr[56:32]
        | 0x80000000u;                               // type=2 ("image") at [127:126]

  // group1 (256 bits) assembled as 4 quadwords, then split into 8 dwords.
  const uint64_t td0 = elems_avail;                  // tensor_dim0  [79:48]
  const uint64_t q0  = 0x20000ull                    // data_size=2 (4 bytes) [17:16]
                     | ((td0 & 0xFFFFull) << 48);
  const uint64_t q1  = (td0 >> 16)                   // tensor_dim0 hi
                     | (1ull << 16)                  // tensor_dim1 = 1   [111:80]
                     | ((uint64_t)tile_elems << 48); // tile_dim0        [127:112]
  const uint64_t q2  = (td0 << 32);                  // tile_dim1=0, tile_dim2=0,
                                                     // tensor_dim0_stride lo [191:160]
  const uint64_t q3  = (td0 >> 32) & 0xFFFFull;      // stride0 hi; tensor_dim1_stride=0

  u32x8 g1;
  g1[0] = (uint32_t)q0;  g1[1] = (uint32_t)(q0 >> 32);
  g1[2] = (uint32_t)q1;  g1[3] = (uint32_t)(q1 >> 32);
  g1[4] = (uint32_t)q2;  g1[5] = (uint32_t)(q2 >> 32);
  g1[6] = (uint32_t)q3;  g1[7] = (uint32_t)(q3 >> 32);

  asm volatile("tensor_load_to_lds %0, %1" :: "s"(g0), "s"(g1) : "memory");
}

// x0 = concat(user_emb, item_emb); acc = x0  (acc lives in d_out)
__global__ void k_init(const float* __restrict__ ue, const float* __restrict__ ie,
                       long nU4, long nTot4,
                       float* __restrict__ x0, float* __restrict__ acc) {
  long i = (long)blockIdx.x * blockDim.x + threadIdx.x;
  if (i >= nTot4) return;
  const float4* src = (i < nU4) ? ((const float4*)ue + i) : ((const float4*)ie + (i - nU4));
  float4 v = *src;
  ((float4*)x0)[i]  = v;
  ((float4*)acc)[i] = v;
}

// deg[col[e]] += 1  (deg pre-zeroed)
__global__ void k_deg(const int* __restrict__ col, float* __restrict__ deg, int E) {
  int e = blockIdx.x * blockDim.x + threadIdx.x;
  if (e < E) atomic_fadd(&deg[col[e]], 1.0f);
}

// deg -> dinv in place: deg>0 ? rsqrt(max(deg,1)) : 0
__global__ void k_dinv(float* __restrict__ deg, int N) {
  int i = blockIdx.x * blockDim.x + threadIdx.x;
  if (i < N) {
    float d = deg[i];
    deg[i] = (d > 0.0f) ? rsqrtf(fmaxf(d, 1.0f)) : 0.0f;
  }
}

// per-edge norm, computed once and reused for all 3 layers
__global__ void k_norm(const int* __restrict__ row, const int* __restrict__ col,
                       const float* __restrict__ dinv, float* __restrict__ nrm, int E) {
  int e = blockIdx.x * blockDim.x + threadIdx.x;
  if (e < E) nrm[e] = dinv[row[e]] * dinv[col[e]];
}

// y[col] += nrm * x[row]  -- one 256-edge tile per block.
// Wave 0 DMAs the tile's (row,col,norm) strips into LDS with three
// tensor_load_to_lds ops, waits TENSORcnt==0, then the barrier releases all
// 8 waves; each wave handles 32 edges, its 32 lanes cooperating on the
// 128-float row (float4 gather + 4 hardware f32 atomics).
__global__ void k_conv(const int* __restrict__ row, const int* __restrict__ col,
                       const float* __restrict__ nrm, const float* __restrict__ x,
                       float* __restrict__ y, int E) {
  __shared__ int sTile[3 * TILE];   // [0,256)=row  [256,512)=col  [512,768)=norm

  const int  tid  = threadIdx.x;
  const int  lane = tid & 31;
  const int  wv   = tid >> 5;
  const long base = (long)blockIdx.x * TILE;

  if (wv == 0) {
    const uint32_t ldsBase = (uint32_t)(uintptr_t)&sTile[0];
    const uint32_t avail   = (uint32_t)(E - base);     // base < E by grid sizing
    tdm_load_1d(ldsBase,                 row + base, avail, TILE);
    tdm_load_1d(ldsBase + TILE * 4,      col + base, avail, TILE);
    tdm_load_1d(ldsBase + 2 * TILE * 4,  nrm + base, avail, TILE);
    __builtin_amdgcn_s_wait_tensorcnt(0);
  }
  __syncthreads();

  const int lo     = wv * 32;
  const int navail = (int)min((long)TILE, (long)(E - base));
  const int cnt    = min(32, navail - lo);   // may be <= 0 on the last tile
  const int fo     = lane * 4;               // this lane's float4 within the row

  for (int j = 0; j < cnt; ++j) {
    int jp = j + 4;                          // prefetch the random gather ahead
    if (jp < cnt)
      __builtin_prefetch(x + (size_t)sTile[lo + jp] * EMB + fo, 0, 0);

    const int   r = sTile[lo + j];                       // LDS broadcast reads
    const int   c = sTile[TILE + lo + j];
    const float w = __int_as_float(sTile[2 * TILE + lo + j]);

    float4 v = *(const float4*)(x + (size_t)r * EMB + fo);   // global_load_b128
    float* dst = y + (size_t)c * EMB + fo;
    atomic_fadd(dst + 0, v.x * w);
    atomic_fadd(dst + 1, v.y * w);
    atomic_fadd(dst + 2, v.z * w);
    atomic_fadd(dst + 3, v.w * w);
  }
}

// acc = (acc + x) * s   (float4); s=1 for inner layers, 0.25 for the last
__global__ void k_acc(float* __restrict__ acc, const float* __restrict__ x,
                      long n4, float s) {
  long i = (long)blockIdx.x * blockDim.x + threadIdx.x;
  if (i >= n4) return;
  float4 a = ((float4*)acc)[i];
  float4 b = ((const float4*)x)[i];
  a.x = (a.x + b.x) * s; a.y = (a.y + b.y) * s;
  a.z = (a.z + b.z) * s; a.w = (a.w + b.w) * s;
  ((float4*)acc)[i] = a;
}

extern "C" void kernel_launch(void* const* d_in, const int* in_sizes, int n_in,
                              void* d_out, int out_size, void* d_ws, size_t ws_size,
                              hipStream_t stream) {
  const int*   edge = (const int*)d_in[0];     // [2, E] int32, flat
  const float* ue   = (const float*)d_in[1];   // [numU, 128] f32
  const float* ie   = (const float*)d_in[2];   // [numI, 128] f32

  const int  E   = in_sizes[0] / 2;
  const int  nU  = in_sizes[1] / EMB;
  const int  nI  = in_sizes[2] / EMB;
  const int  N   = nU + nI;
  const long ND  = (long)N * EMB;
  const long n4  = ND / 4;

  const int* row = edge;                       // edge_index[0]
  const int* col = edge + E;                   // edge_index[1]

  // workspace layout: x0 | x1 | dinv(padded) | norm
  char*  ws   = (char*)d_ws;
  float* x0   = (float*)ws;  ws += ND * sizeof(float);
  float* x1   = (float*)ws;  ws += ND * sizeof(float);
  float* dinv = (float*)ws;  ws += ((long)(N + 63) / 64 * 64) * sizeof(float);
  float* nrm  = (float*)ws;
  float* acc  = (float*)d_out;                 // accumulator lives in d_out

  // x0 = concat(user, item); acc = x0
  k_init<<<cdiv(n4, BLK), BLK, 0, stream>>>(ue, ie, (long)nU * EMB / 4, n4, x0, acc);

  // degrees / dinv / per-edge norm (computed once, reused for all layers)
  hipMemsetAsync(dinv, 0, (size_t)N * sizeof(float), stream);
  k_deg <<<cdiv(E, BLK), BLK, 0, stream>>>(col, dinv, E);
  k_dinv<<<cdiv(N, BLK), BLK, 0, stream>>>(dinv, N);
  k_norm<<<cdiv(E, BLK), BLK, 0, stream>>>(row, col, dinv, nrm, E);

  float* xa = x0;
  float* xb = x1;
  for (int l = 0; l < 3; ++l) {
    hipMemsetAsync(xb, 0, (size_t)ND * sizeof(float), stream);
    k_conv<<<cdiv(E, TILE), BLK, 0, stream>>>(row, col, nrm, xa, xb, E);
    k_acc <<<cdiv(n4, BLK), BLK, 0, stream>>>(acc, xb, n4, (l == 2) ? 0.25f : 1.0f);
    float* t = xa; xa = xb; xb = t;
  }
}